// GroupRecombine_77137612636750
// MI455X (gfx1250) — compile-verified
//
#include <hip/hip_runtime.h>
#include <math.h>

#define LEN 256           // 16 x 16 tile per row
#define LDS_STRIDE 20     // 16 + 4 pad: conflict-free for both access patterns
#define WAVES_PER_BLOCK 8

typedef float v2f __attribute__((ext_vector_type(2)));
typedef float v8f __attribute__((ext_vector_type(8)));

// H16[i][j] = (-1)^popcount(i & j)
__device__ __forceinline__ float hval(int i, int j) {
    return (__builtin_popcount(i & j) & 1) ? -1.0f : 1.0f;
}

// Order LDS store->load phases within a wave (DS unit is in-order; the
// memory clobber stops compiler reordering, the wait drains DScnt).
__device__ __forceinline__ void lds_fence() {
    __asm__ volatile("s_wait_dscnt 0" ::: "memory");
}

// One FWHT-256 stage on a row held as B-operand regs b[8] (lane (hi,lo) holds
// X[4c+2hi + t][lo] at b[2c+t]).  Computes W = (H16 * X * H16)^T in C/D layout
// via  Y = H*X  (4 chained WMMA K=4),  LDS transpose,  W = H*Y^T.
__device__ __forceinline__ v8f fwht_stage(const float* b, float* lds, int lo, int hi) {
    v8f acc = {};
#pragma unroll
    for (int c = 0; c < 4; ++c) {
        v2f a, bb;
        const int col = 4 * c + 2 * hi;
        a.x = hval(lo, col);
        a.y = hval(lo, col + 1);
        bb.x = b[2 * c];
        bb.y = b[2 * c + 1];
        acc = __builtin_amdgcn_wmma_f32_16x16x4_f32(false, a, false, bb,
                                                    (short)0, acc, false, false);
    }
    // Y (C/D layout: vgpr k -> row k+8*hi, col lo) -> LDS, natural order
#pragma unroll
    for (int k = 0; k < 8; ++k)
        lds[(k + 8 * hi) * LDS_STRIDE + lo] = acc[k];
    lds_fence();
    // read back transposed into B layout: bT[2c+t] = Y[lo][4c+2hi+t]
    float bt[8];
#pragma unroll
    for (int c = 0; c < 4; ++c) {
        bt[2 * c]     = lds[lo * LDS_STRIDE + 4 * c + 2 * hi];
        bt[2 * c + 1] = lds[lo * LDS_STRIDE + 4 * c + 2 * hi + 1];
    }
    lds_fence();
    v8f w = {};
#pragma unroll
    for (int c = 0; c < 4; ++c) {
        v2f a, bb;
        const int col = 4 * c + 2 * hi;
        a.x = hval(lo, col);
        a.y = hval(lo, col + 1);
        bb.x = bt[2 * c];
        bb.y = bt[2 * c + 1];
        w = __builtin_amdgcn_wmma_f32_16x16x4_f32(false, a, false, bb,
                                                  (short)0, w, false, false);
    }
    return w;
}

// log_softmax over the 256-element row spread as 8 values/lane; writes
// v[i] - logsumexp to outrow[idx[i]].
__device__ __forceinline__ void logsoftmax_store(const float* v, float* outrow,
                                                 const int* idx) {
    float m = v[0];
#pragma unroll
    for (int i = 1; i < 8; ++i) m = fmaxf(m, v[i]);
#pragma unroll
    for (int off = 16; off >= 1; off >>= 1) m = fmaxf(m, __shfl_xor(m, off, 32));
    float s = 0.0f;
#pragma unroll
    for (int i = 0; i < 8; ++i) s += __expf(v[i] - m);
#pragma unroll
    for (int off = 16; off >= 1; off >>= 1) s += __shfl_xor(s, off, 32);
    const float lse = m + __logf(s);
#pragma unroll
    for (int i = 0; i < 8; ++i) outrow[idx[i]] = v[i] - lse;
}

__global__ void __launch_bounds__(WAVES_PER_BLOCK * 32)
wh_convolve_kernel(const float* __restrict__ g0, const float* __restrict__ g1,
                   const float* __restrict__ g2, float* __restrict__ out, int batch) {
    __shared__ float ldsbuf[WAVES_PER_BLOCK * 16 * LDS_STRIDE];
    const int lane = threadIdx.x & 31;
    const int wib  = threadIdx.x >> 5;
    float* lds = &ldsbuf[wib * 16 * LDS_STRIDE];
    const int lo = lane & 15, hi = lane >> 4;

    const int waveId = blockIdx.x * WAVES_PER_BLOCK + wib;
    const int nWaves = gridDim.x * WAVES_PER_BLOCK;

    // per-lane vector indices for the B-operand load layout
    int idx[8];
#pragma unroll
    for (int c = 0; c < 4; ++c) {
        idx[2 * c]     = 64 * c + 32 * hi + lo;
        idx[2 * c + 1] = 64 * c + 32 * hi + 16 + lo;
    }
    const float s1 = exp2f(-8.0f / 3.0f / 2.0f);  // per-share WH scale
    const float s2 = 1.0f / 16.0f;                // final WH scale (n_shares=1)
    const size_t outStride = (size_t)batch * LEN;

    for (int row = waveId; row < batch; row += nWaves) {
        const float* ins[3] = { g0 + (size_t)row * LEN,
                                g1 + (size_t)row * LEN,
                                g2 + (size_t)row * LEN };
        v8f prod = {};
#pragma unroll
        for (int t = 0; t < 3; ++t) {
            float x[8];
#pragma unroll
            for (int i = 0; i < 8; ++i) x[i] = ins[t][idx[i]];
            // log_softmax of the raw input -> output section t
            logsoftmax_store(x, out + (size_t)t * outStride + (size_t)row * LEN, idx);
            v8f w = fwht_stage(x, lds, lo, hi);
            if (t == 0) {
#pragma unroll
                for (int k = 0; k < 8; ++k) prod[k] = w[k] * s1;
            } else {
#pragma unroll
                for (int k = 0; k < 8; ++k) prod[k] *= w[k] * s1;
            }
        }
        // prod (C/D layout) -> LDS natural -> B layout for second FWHT
#pragma unroll
        for (int k = 0; k < 8; ++k)
            lds[(k + 8 * hi) * LDS_STRIDE + lo] = prod[k];
        lds_fence();
        float pb[8];
#pragma unroll
        for (int c = 0; c < 4; ++c) {
            pb[2 * c]     = lds[(4 * c + 2 * hi) * LDS_STRIDE + lo];
            pb[2 * c + 1] = lds[(4 * c + 2 * hi + 1) * LDS_STRIDE + lo];
        }
        lds_fence();
        v8f res = fwht_stage(pb, lds, lo, hi);
        float r[8];
        int ridx[8];
#pragma unroll
        for (int k = 0; k < 8; ++k) {
            r[k] = res[k] * s2;
            ridx[k] = (k + 8 * hi) * 16 + lo;   // C/D layout -> natural index
        }
        logsoftmax_store(r, out + 3 * outStride + (size_t)row * LEN, ridx);
    }
}

extern "C" void kernel_launch(void* const* d_in, const int* in_sizes, int n_in,
                              void* d_out, int out_size, void* d_ws, size_t ws_size,
                              hipStream_t stream) {
    (void)n_in; (void)d_ws; (void)ws_size; (void)out_size;
    const float* b0 = (const float*)d_in[0];
    const float* b1 = (const float*)d_in[1];
    const float* mk = (const float*)d_in[2];
    float* out = (float*)d_out;
    const int batch = in_sizes[0] / LEN;

    // ~4 rows per wave at BATCH=131072: 4096 blocks x 8 waves
    int blocks = (batch + 31) / 32;
    if (blocks < 1) blocks = 1;
    if (blocks > 65535) blocks = 65535;
    wh_convolve_kernel<<<blocks, WAVES_PER_BLOCK * 32, 0, stream>>>(b0, b1, mk, out, batch);
}